// GlobalAttentionPooling_81724637708861
// MI455X (gfx1250) — compile-verified
//
#include <hip/hip_runtime.h>
#include <hip/hip_bf16.h>
#include <math.h>

typedef __attribute__((ext_vector_type(2))) float v2f;
typedef __attribute__((ext_vector_type(8))) float v8f;

// ---------------------------------------------------------------------------
// Kernel 1: segment offsets via binary search (batch is sorted, values in [0,G))
// off[g] = first i with batch[i] >= g ; off[G] = N
// ---------------------------------------------------------------------------
__global__ void seg_offsets_kernel(const int* __restrict__ batch, int* __restrict__ off,
                                   int N, int G) {
  int g = blockIdx.x * blockDim.x + threadIdx.x;
  if (g > G) return;
  int lo = 0, hi = N;
  while (lo < hi) {
    int mid = (lo + hi) >> 1;
    if (batch[mid] < g) lo = mid + 1; else hi = mid;
  }
  off[g] = lo;
}

// ---------------------------------------------------------------------------
// Kernel 2: gate[i] = dot(x[i,:], W) + b  via fp32 WMMA 16x16x4.
// One wave handles a 16-row tile of x. Per CDNA5 ISA A-layout (16x4 fp32):
//   lanes 0-15  : row M = lane,    VGPR0/1 = K0,K1
//   lanes 16-31 : row M = lane-16, VGPR0/1 = K2,K3
// B trick: broadcast W[k..k+3] into ALL 16 columns of B (every lane supplies the
// same per-K W value).  Then every column of D computes the identical dot
// product -> column 0 is exact, and the B operand is a plain unconditional
// global_load_b64 with no exec-mask predication in the inner loop.
// D col 0 sits in lane 0 (rows 0-7 in c[0..7]) and lane 16 (rows 8-15).
// ---------------------------------------------------------------------------
__global__ void gate_wmma_kernel(const float* __restrict__ x, const float* __restrict__ W,
                                 const float* __restrict__ b, float* __restrict__ gate,
                                 int N, int D) {
  const int lane          = threadIdx.x & 31;
  const int wave          = threadIdx.x >> 5;
  const int wavesPerBlock = blockDim.x >> 5;
  const int numTiles      = (N + 15) >> 4;
  const int stride        = gridDim.x * wavesPerBlock;
  const float bias        = b[0];

  const int half = lane >> 4;    // 0: lanes 0-15, 1: lanes 16-31
  const int mrow = lane & 15;    // A-matrix row within the tile
  const int koff = half << 1;    // K offset supplied by this half-wave (0 or 2)

  for (int tile = blockIdx.x * wavesPerBlock + wave; tile < numTiles; tile += stride) {
    int row  = tile * 16 + mrow;
    int rowC = row < N ? row : N - 1;          // clamp: EXEC must stay all-ones for WMMA
    const float* xrow = x + (size_t)rowC * (size_t)D + koff;
    const float* wptr = W + koff;

    v8f c = {0.0f, 0.0f, 0.0f, 0.0f, 0.0f, 0.0f, 0.0f, 0.0f};
    for (int k = 0; k < D; k += 4) {
      v2f a  = *(const v2f*)(xrow + k);        // A[m, K+koff .. K+koff+1]
      v2f bb = *(const v2f*)(wptr + k);        // W broadcast into every B column
      c = __builtin_amdgcn_wmma_f32_16x16x4_f32(false, a, false, bb, (short)0, c,
                                                false, false);
    }

    if (mrow == 0) {                           // lanes 0 and 16 hold D column 0
      int rbase = tile * 16 + (half << 3);     // lane 0 -> rows 0-7, lane 16 -> rows 8-15
      #pragma unroll
      for (int j = 0; j < 8; ++j) {
        int r = rbase + j;
        if (r < N) gate[r] = c[j] + bias;
      }
    }
  }
}

// ---------------------------------------------------------------------------
// Kernel 3: per-graph stable segment softmax + weighted pooling.
// grid = (G, ceil(D/256)), block = 256. Each thread owns one output column.
// Alphas for 256-row chunks are staged in LDS so each exp is computed once.
// ---------------------------------------------------------------------------
__global__ void pool_kernel(const float* __restrict__ x, const float* __restrict__ gate,
                            const int* __restrict__ off, float* __restrict__ out, int D) {
  const int g   = blockIdx.x;
  const int s   = off[g];
  const int e   = off[g + 1];
  const int tid = threadIdx.x;
  const int col = blockIdx.y * 256 + tid;

  __shared__ float red[256];
  __shared__ float aLds[256];

  // segment max
  float m = -INFINITY;
  for (int r = s + tid; r < e; r += 256) m = fmaxf(m, gate[r]);
  red[tid] = m;
  __syncthreads();
  for (int w = 128; w > 0; w >>= 1) {
    if (tid < w) red[tid] = fmaxf(red[tid], red[tid + w]);
    __syncthreads();
  }
  m = red[0];
  __syncthreads();

  // segment sum of exp(gate - m)
  float lsum = 0.0f;
  for (int r = s + tid; r < e; r += 256) lsum += __expf(gate[r] - m);
  red[tid] = lsum;
  __syncthreads();
  for (int w = 128; w > 0; w >>= 1) {
    if (tid < w) red[tid] += red[tid + w];
    __syncthreads();
  }
  float denom = red[0];
  float inv   = denom > 0.0f ? 1.0f / denom : 0.0f;  // empty segment -> loop below is empty
  __syncthreads();

  // weighted pooling: stream rows, alphas staged per 256-row chunk
  float acc = 0.0f;
  for (int base = s; base < e; base += 256) {
    int cnt = e - base;
    if (cnt > 256) cnt = 256;
    if (tid < cnt) aLds[tid] = __expf(gate[base + tid] - m) * inv;
    __syncthreads();
    if (col < D) {
      const float* xr = x + (size_t)base * (size_t)D + col;
      for (int j = 0; j < cnt; ++j) {
        acc = fmaf(aLds[j], xr[(size_t)j * (size_t)D], acc);
      }
    }
    __syncthreads();
  }
  if (col < D) out[(size_t)g * (size_t)D + col] = acc;
}

// ---------------------------------------------------------------------------
extern "C" void kernel_launch(void* const* d_in, const int* in_sizes, int n_in,
                              void* d_out, int out_size, void* d_ws, size_t ws_size,
                              hipStream_t stream) {
  (void)n_in; (void)ws_size;
  const float* x     = (const float*)d_in[0];
  const float* W     = (const float*)d_in[1];
  const float* b     = (const float*)d_in[2];
  const int*   batch = (const int*)d_in[3];   // sorted graph ids

  const int D = in_sizes[1];                  // W is [D,1]
  const int N = in_sizes[3];                  // batch is [N]
  const int G = out_size / D;                 // out is [G,D]

  int*   off  = (int*)d_ws;                          // (G+1) ints
  float* gate = (float*)((char*)d_ws + 4096);        // N floats
  float* out  = (float*)d_out;                       // [G, D] fp32

  seg_offsets_kernel<<<(G + 1 + 255) / 256, 256, 0, stream>>>(batch, off, N, G);

  const int wavesPerBlock = 8;                        // 256 threads = 8 wave32
  const int numTiles      = (N + 15) / 16;
  int gateBlocks = (numTiles + wavesPerBlock - 1) / wavesPerBlock;
  if (gateBlocks > 4096) gateBlocks = 4096;
  gate_wmma_kernel<<<gateBlocks, 256, 0, stream>>>(x, W, b, gate, N, D);

  dim3 grid((unsigned)G, (unsigned)((D + 255) / 256));
  pool_kernel<<<grid, 256, 0, stream>>>(x, gate, off, out, D);
}